// NormLayer_46136538694219
// MI455X (gfx1250) — compile-verified
//
#include <hip/hip_runtime.h>
#include <math.h>

// Instance-norm over (N=64, C=3, H=512, W=512) fp32.
// One 1024-thread block per (n,c) channel; 192 blocks total.
// Pass 1: async global->LDS streaming (8-buffer ring, 7 in flight) + WMMA-assisted
//         sum/sumsq reduction. Pass 2: normalize; input re-read hits L2, output
//         stored non-temporally to preserve L2 for input data.

#define HWN   262144              // 512*512 elements per channel
#define BLOCK 1024                // 32 waves
#define ITERS (HWN / (BLOCK * 4)) // 64 float4 tiles per thread
#define NBUF  8                   // async ring depth (128 KB LDS stage)
#define DEPTH 7                   // async copies in flight per wave

typedef __attribute__((ext_vector_type(2))) float v2f;
typedef __attribute__((ext_vector_type(4))) float v4f;
typedef __attribute__((ext_vector_type(8))) float v8f;
typedef __attribute__((ext_vector_type(4))) int   v4i;

#if defined(__gfx1250__) && __has_builtin(__builtin_amdgcn_global_load_async_to_lds_b128)
#define HAS_ASYNC 1
#else
#define HAS_ASYNC 0
#endif

#if defined(__gfx1250__) && __has_builtin(__builtin_amdgcn_wmma_f32_16x16x4_f32)
#define HAS_WMMA_F32X4 1
#else
#define HAS_WMMA_F32X4 0
#endif

__device__ __forceinline__ void async_copy16(const v4f* g, v4f* l) {
#if HAS_ASYNC
  // Builtin signature (per clang diagnostic): (v4i AS1*, v4i AS3*, Ii, Ii)
  v4i* gg = reinterpret_cast<v4i*>(const_cast<v4f*>(g));
  v4i* ll = reinterpret_cast<v4i*>(l);
  __builtin_amdgcn_global_load_async_to_lds_b128(
      (__attribute__((address_space(1))) v4i*)gg,
      (__attribute__((address_space(3))) v4i*)ll,
      /*offset=*/0, /*cpol=*/0);
#else
  *l = *g;
#endif
}

template <int N>
__device__ __forceinline__ void wait_async() {
#if HAS_ASYNC
  asm volatile("s_wait_asynccnt %0" ::"n"(N) : "memory");
#endif
}

__device__ __forceinline__ void wait_async_rt(int n) {
  switch (n) {
    case 0: wait_async<0>(); break;
    case 1: wait_async<1>(); break;
    case 2: wait_async<2>(); break;
    case 3: wait_async<3>(); break;
    case 4: wait_async<4>(); break;
    case 5: wait_async<5>(); break;
    default: wait_async<6>(); break;
  }
}

// Full 32-lane sum, result valid in every lane.
__device__ __forceinline__ float wave_sum(float v) {
#if HAS_WMMA_F32X4
  // A (16x4 f32): VGPR0 = K0 (lanes 0-15 -> M=lane) / K2 (lanes 16-31 -> M=lane-16),
  // VGPR1 = K1/K3 (zeroed). B = all ones -> D[m][n] = v_m + v_{m+16} (layout-proof).
  v2f a; a.x = v;    a.y = 0.0f;
  v2f b; b.x = 1.0f; b.y = 1.0f;
  v8f c = {0.f, 0.f, 0.f, 0.f, 0.f, 0.f, 0.f, 0.f};
  v8f d = __builtin_amdgcn_wmma_f32_16x16x4_f32(
      /*neg_a=*/false, a, /*neg_b=*/false, b,
      /*c_mod=*/(short)0, c, /*reuse_a=*/false, /*reuse_b=*/false);
  // lanes 0-15 hold rows M=0..7, lanes 16-31 hold rows M=8..15
  float r = ((d[0] + d[1]) + (d[2] + d[3])) + ((d[4] + d[5]) + (d[6] + d[7]));
  r += __shfl_xor(r, 16, 32);
  return r;
#else
  for (int off = 16; off > 0; off >>= 1) v += __shfl_xor(v, off, 32);
  return v;
#endif
}

__global__ __launch_bounds__(BLOCK) void inorm_kernel(
    const float* __restrict__ in, float* __restrict__ out) {
  __shared__ v4f    stage[NBUF][BLOCK];   // 128 KB async staging ring
  __shared__ float2 wred[BLOCK / 32];     // per-wave partials
  __shared__ float  stats[2];             // mean, inv_std

  const int tid  = threadIdx.x;
  const int lane = tid & 31;
  const int wave = tid >> 5;

  const v4f* gsrc = (const v4f*)(in + (size_t)blockIdx.x * HWN);
  v4f*       gdst = (v4f*)(out + (size_t)blockIdx.x * HWN);

  float s = 0.0f, q = 0.0f;

  auto consume = [&](int i) {
    v4f xv = stage[i & (NBUF - 1)][tid];
    s += (xv.x + xv.y) + (xv.z + xv.w);
    q += (xv.x * xv.x + xv.y * xv.y) + (xv.z * xv.z + xv.w * xv.w);
  };

  // ---- Pass 1: async-stream input through LDS, accumulate sum / sumsq ----
  // Each wave only writes/reads its own 32 lanes' slots: no barriers needed.
  // 7 async copies in flight per wave -> 32 waves * 7 * 512B = 112 KB/WGP MLP.
#pragma unroll
  for (int p = 0; p < DEPTH; ++p)
    async_copy16(gsrc + (size_t)p * BLOCK + tid, &stage[p][tid]);

  for (int i = 0; i < ITERS - DEPTH; ++i) {
    async_copy16(gsrc + (size_t)(i + DEPTH) * BLOCK + tid,
                 &stage[(i + DEPTH) & (NBUF - 1)][tid]);
    wait_async<DEPTH>();  // oldest in-flight tile (i) is complete
    consume(i);
  }
#pragma unroll
  for (int k = DEPTH - 1; k >= 0; --k) {  // drain: thresholds 6,5,...,0
    wait_async_rt(k);
    consume(ITERS - 1 - k);
  }

  // ---- Block reduction: wave (WMMA) -> LDS -> wave0 (WMMA) ----
  float S = wave_sum(s);
  float Q = wave_sum(q);
  if (lane == 0) wred[wave] = make_float2(S, Q);
  __syncthreads();

  if (wave == 0) {
    float2 p  = wred[lane];  // exactly 32 wave partials
    float  ss = wave_sum(p.x);
    float  qq = wave_sum(p.y);
    if (lane == 0) {
      const float hw   = (float)HWN;
      float       mean = ss / hw;
      float       var  = (qq - ss * ss / hw) * (1.0f / (hw - 1.0f));
      var              = fmaxf(var, 0.0f);
      stats[0]         = mean;
      stats[1]         = 1.0f / (sqrtf(var) + 1e-8f);
    }
  }
  __syncthreads();

  const float mean = stats[0];
  const float inv  = stats[1];

  // ---- Pass 2: normalize. Input re-read hits L2 (just streamed by this WGP);
  // output is write-once -> non-temporal store to keep L2 for input data. ----
#pragma unroll 8
  for (int i = 0; i < ITERS; ++i) {
    const size_t idx = (size_t)i * BLOCK + tid;
    v4f          xv  = gsrc[idx];
    v4f          yv;
    yv.x = (xv.x - mean) * inv;
    yv.y = (xv.y - mean) * inv;
    yv.z = (xv.z - mean) * inv;
    yv.w = (xv.w - mean) * inv;
    __builtin_nontemporal_store(yv, &gdst[idx]);
  }
}

extern "C" void kernel_launch(void* const* d_in, const int* in_sizes, int n_in,
                              void* d_out, int out_size, void* d_ws, size_t ws_size,
                              hipStream_t stream) {
  const float* x  = (const float*)d_in[0];
  float*       y  = (float*)d_out;
  const int    nc = in_sizes[0] / HWN;  // 64*3 = 192 channels
  inorm_kernel<<<nc, BLOCK, 0, stream>>>(x, y);
}